// VGCNLayer_32306744000543
// MI455X (gfx1250) — compile-verified
//
#include <hip/hip_runtime.h>
#include <hip/hip_bf16.h>

typedef float v2f __attribute__((ext_vector_type(2)));
typedef float v4f __attribute__((ext_vector_type(4)));
typedef float v8f __attribute__((ext_vector_type(8)));

#define ALPHA 0.5f
#define DFEAT 128

// ---------------- zero workspace (agg + deg) ----------------
__global__ void vgcn_zero_kernel(float* __restrict__ p, long n) {
    long i = (long)blockIdx.x * blockDim.x + threadIdx.x;
    if (i < n) p[i] = 0.0f;
}

// ---------------- in-degree via float atomics ----------------
__global__ void vgcn_degree_kernel(const int* __restrict__ dst,
                                   float* __restrict__ deg, int E) {
    int e = blockIdx.x * blockDim.x + threadIdx.x;
    if (e < E) {
        __hip_atomic_fetch_add(&deg[dst[e]], 1.0f,
                               __ATOMIC_RELAXED, __HIP_MEMORY_SCOPE_AGENT);
    }
}

// ---------------- h = (F @ W^T) * deg^-1/2 via fp32 WMMA ----------------
// One wave -> one 16x16 tile of h. Block = 8 waves = 16 rows x 128 cols.
// A(MxK)=F[rows,k]; B(KxN)=W^T[k,d]=W[d,k]. K accumulated 4 at a time.
__global__ __launch_bounds__(256)
void vgcn_gemm_norm_kernel(const float* __restrict__ F,
                           const float* __restrict__ W,
                           const float* __restrict__ deg,
                           float* __restrict__ hlin, int N) {
    const int lane  = threadIdx.x & 31;
    const int wave  = threadIdx.x >> 5;
    const int row0  = blockIdx.x * 16;      // M tile
    const int col0  = wave * 16;            // N tile (8 waves cover D=128)
    const int m     = lane & 15;            // M (for A) / N (for B) within tile
    const int khalf = (lane >> 4) << 1;     // lanes 0-15 -> K=0,1 ; lanes 16-31 -> K=2,3

    // Clamp row for safety (N is a multiple of 16 here, keeps EXEC all-ones).
    const int arow_i = (row0 + m < N) ? (row0 + m) : (N - 1);
    const float* __restrict__ arow = F + (long)arow_i * DFEAT + khalf;
    const float* __restrict__ brow = W + (long)(col0 + m) * DFEAT + khalf;

    v8f c = {0.f, 0.f, 0.f, 0.f, 0.f, 0.f, 0.f, 0.f};
#pragma unroll
    for (int k = 0; k < DFEAT; k += 4) {
        v2f a = *(const v2f*)(arow + k);   // A: {K=khalf, K=khalf+1} of row M
        v2f b = *(const v2f*)(brow + k);   // B: {K=khalf, K=khalf+1} of col N
        c = __builtin_amdgcn_wmma_f32_16x16x4_f32(
                /*neg_a=*/false, a, /*neg_b=*/false, b,
                /*c_mod=*/(short)0, c, /*reuse_a=*/false, /*reuse_b=*/false);
    }

    // C/D layout: VGPR r -> M=r (lanes 0-15) / M=r+8 (lanes 16-31), N=lane&15.
    const int nout = lane & 15;
    const int mhi  = (lane >> 4) << 3;
#pragma unroll
    for (int r = 0; r < 8; ++r) {
        int row = row0 + mhi + r;
        if (row < N) {
            float nrm = __frsqrt_rn(fmaxf(deg[row], 1.0f));
            hlin[(long)row * DFEAT + col0 + nout] = c[r] * nrm;
        }
    }
}

// ---------------- edge scatter: agg[dst] += hlin[src] ----------------
// One wave32 per edge; lane handles 4 consecutive floats (float4 load,
// 4x global_atomic_add_f32). 512B per edge, fully coalesced; h and agg
// both fit in the 192MB L2 so this stays on-chip.
__global__ __launch_bounds__(256)
void vgcn_scatter_kernel(const int* __restrict__ src,
                         const int* __restrict__ dst,
                         const float* __restrict__ hlin,
                         float* __restrict__ agg, int E) {
    const int lane = threadIdx.x & 31;
    const int e    = blockIdx.x * 8 + (threadIdx.x >> 5);
    if (e >= E) return;
    const long soff = (long)src[e] * DFEAT + lane * 4;
    const long doff = (long)dst[e] * DFEAT + lane * 4;
    v4f v = *(const v4f*)(hlin + soff);
    float* p = agg + doff;
    __hip_atomic_fetch_add(p + 0, v.x, __ATOMIC_RELAXED, __HIP_MEMORY_SCOPE_AGENT);
    __hip_atomic_fetch_add(p + 1, v.y, __ATOMIC_RELAXED, __HIP_MEMORY_SCOPE_AGENT);
    __hip_atomic_fetch_add(p + 2, v.z, __ATOMIC_RELAXED, __HIP_MEMORY_SCOPE_AGENT);
    __hip_atomic_fetch_add(p + 3, v.w, __ATOMIC_RELAXED, __HIP_MEMORY_SCOPE_AGENT);
}

// ---------------- finalize: relu(a*norm*agg + a*ri/deg + (1-a)*F) ----------------
__global__ void vgcn_finalize_kernel(const float* __restrict__ agg,
                                     const float* __restrict__ deg,
                                     const float* __restrict__ F,
                                     const float* __restrict__ Ri,
                                     float* __restrict__ out, long total) {
    long i = (long)blockIdx.x * blockDim.x + threadIdx.x;
    if (i >= total) return;
    long n = i >> 7;  // /128
    float dg  = fmaxf(deg[n], 1.0f);
    float nrm = __frsqrt_rn(dg);
    float v = ALPHA * nrm * agg[i] + ALPHA * Ri[i] * (1.0f / dg)
            + (1.0f - ALPHA) * F[i];
    out[i] = fmaxf(v, 0.0f);
}

extern "C" void kernel_launch(void* const* d_in, const int* in_sizes, int n_in,
                              void* d_out, int out_size, void* d_ws, size_t ws_size,
                              hipStream_t stream) {
    const float* F   = (const float*)d_in[0];   // features [N,128]
    const float* Ri  = (const float*)d_in[1];   // initial_features [N,128]
    const float* W   = (const float*)d_in[2];   // W [128,128]
    const int*   src = (const int*)d_in[3];     // [E]
    const int*   dst = (const int*)d_in[4];     // [E]

    const int N = in_sizes[0] / DFEAT;
    const int E = in_sizes[3];

    float* out = (float*)d_out;                 // reused as hlin scratch, then final
    float* agg = (float*)d_ws;                  // [N*128]
    float* deg = agg + (size_t)N * DFEAT;       // [N]

    // 1) zero agg + deg
    long zn = (long)N * DFEAT + N;
    vgcn_zero_kernel<<<(unsigned)((zn + 255) / 256), 256, 0, stream>>>(agg, zn);

    // 2) in-degree
    vgcn_degree_kernel<<<(E + 255) / 256, 256, 0, stream>>>(dst, deg, E);

    // 3) WMMA GEMM + src-side norm -> hlin (in d_out)
    vgcn_gemm_norm_kernel<<<(N + 15) / 16, 256, 0, stream>>>(F, W, deg, out, N);

    // 4) edge scatter into agg
    vgcn_scatter_kernel<<<(E + 7) / 8, 256, 0, stream>>>(src, dst, out, agg, E);

    // 5) finalize into d_out
    long total = (long)N * DFEAT;
    vgcn_finalize_kernel<<<(unsigned)((total + 255) / 256), 256, 0, stream>>>(
        agg, deg, F, Ri, out, total);
}